// STVMLapShot_33406255629075
// MI455X (gfx1250) — compile-verified
//
#include <hip/hip_runtime.h>
#include <hip/hip_bf16.h>
#include <math.h>

// ---------------------------------------------------------------------------
// STVMLapShot on gfx1250 (MI455X).
// fp32 WMMA (v_wmma_f32_16x16x4_f32) for both Gram GEMMs; everything else is
// small/latency-bound. All grids fixed at MAX_N for graph capture; device-side
// nAll gates the work. Requires ws_size >= ~560 MB.
// ---------------------------------------------------------------------------

#define D_FEAT 640
#define CMAX   8
#define MAX_N  8192
#define KMAX   16
#define LDQ    320
#define PROTO_FLAG 0x40000000

typedef float v2f __attribute__((ext_vector_type(2)));
typedef float v8f __attribute__((ext_vector_type(8)));

struct Scalars {
  int C;
  int nAll;
  int kVal;
  int done;
  unsigned diffBits;
  float lamDen;
};
struct IPair { int a, b; };

// ------------------------------ prototypes ---------------------------------
__global__ __launch_bounds__(256)
void k_protos(const float* __restrict__ fs, const int* __restrict__ ys,
              float* __restrict__ protos, Scalars* sc, int S) {
  __shared__ float cnt[CMAX];
  __shared__ int labels[64];
  int tid = threadIdx.x;
  if (S > 64) S = 64;
  if (tid == 0) {
    int mx = 0;
    for (int s = 0; s < S; ++s) {
      int l = ys[s]; if (l < 0) l = 0; if (l >= CMAX) l = CMAX - 1;
      labels[s] = l; if (l > mx) mx = l;
    }
    sc->C = mx + 1;
    for (int c = 0; c < CMAX; ++c) cnt[c] = 0.f;
    for (int s = 0; s < S; ++s) cnt[labels[s]] += 1.f;
  }
  __syncthreads();
  for (int d = tid; d < D_FEAT; d += 256) {
    float acc[CMAX];
    for (int c = 0; c < CMAX; ++c) acc[c] = 0.f;
    for (int s = 0; s < S; ++s) acc[labels[s]] += fs[s * D_FEAT + d];
    for (int c = 0; c < CMAX; ++c)
      protos[c * D_FEAT + d] = (cnt[c] > 0.f) ? acc[c] / cnt[c] : 0.f;
  }
}

// --------------------- query->prototype distances + y_boot -----------------
__global__ __launch_bounds__(256)
void k_distq(const float* __restrict__ fq, const float* __restrict__ protos,
             const Scalars* __restrict__ sc, float* __restrict__ distq,
             int* __restrict__ yboot) {
  __shared__ float sred[256][CMAX];
  int q = blockIdx.x, tid = threadIdx.x;
  int C = sc->C;
  float acc[CMAX];
  for (int c = 0; c < CMAX; ++c) acc[c] = 0.f;
  for (int d = tid; d < D_FEAT; d += 256) {
    float qv = fq[q * D_FEAT + d];
    for (int c = 0; c < C; ++c) {
      float df = qv - protos[c * D_FEAT + d];
      acc[c] += df * df;
    }
  }
  for (int c = 0; c < CMAX; ++c) sred[tid][c] = acc[c];
  __syncthreads();
  for (int s = 128; s > 0; s >>= 1) {
    if (tid < s) for (int c = 0; c < CMAX; ++c) sred[tid][c] += sred[tid + s][c];
    __syncthreads();
  }
  if (tid == 0) {
    float best = 1e30f; int bi = 0;
    for (int c = 0; c < C; ++c) {
      float dd = sqrtf(fmaxf(sred[0][c], 1e-30f));
      distq[q * CMAX + c] = dd;
      if (dd < best) { best = dd; bi = c; }
    }
    yboot[q] = bi;
  }
}

// --------------------------- fp32 WMMA Gram kernel -------------------------
// G[0:n,0:n] = X @ X^T, X row stride = D_FEAT.  Block = 128 threads (4 waves),
// each wave computes one 16x16 tile of a 32x32 block tile via
// v_wmma_f32_16x16x4_f32, K panels of 64 staged through LDS.
__global__ __launch_bounds__(128)
void k_gram_wmma(const float* __restrict__ X, float* __restrict__ G,
                 const Scalars* __restrict__ sc, int nFixed, int ldG) {
  int n = sc ? sc->nAll : nFixed;
  int rowBase = blockIdx.y * 32, colBase = blockIdx.x * 32;
  if (rowBase >= n || colBase >= n) return;

  __shared__ float As[32][68];   // stride 68 floats: 16B-aligned float4 slots
  __shared__ float Bs[32][68];

  int tid = threadIdx.x;
  int wave = tid >> 5, lane = tid & 31;
  int mi = lane & 15, hi = lane >> 4;
  int wr = (wave >> 1) * 16, wc = (wave & 1) * 16;

  v8f acc = {};

  for (int k0 = 0; k0 < D_FEAT; k0 += 64) {
    // stage A (rows rowBase..+31) and B (rows colBase..+31), 64 K-columns
    for (int it = 0; it < 4; ++it) {
      int idx = tid + it * 128;          // 0..511 float4 slots (32 rows x 16)
      int r = idx >> 4;
      int c4 = (idx & 15) * 4;
      float4 va = {0.f, 0.f, 0.f, 0.f};
      float4 vb = {0.f, 0.f, 0.f, 0.f};
      int ga = rowBase + r, gb = colBase + r;
      if (ga < n) va = *(const float4*)(X + (size_t)ga * D_FEAT + k0 + c4);
      if (gb < n) vb = *(const float4*)(X + (size_t)gb * D_FEAT + k0 + c4);
      *(float4*)&As[r][c4] = va;
      *(float4*)&Bs[r][c4] = vb;
    }
    if (k0 + 64 < D_FEAT) {
      int ga = rowBase + (tid >> 2);
      if (ga < n) __builtin_prefetch(X + (size_t)ga * D_FEAT + k0 + 64, 0, 1);
    }
    __syncthreads();

    for (int kc = 0; kc < 64; kc += 4) {
      // A frag (16x4 f32): lanes0-15 K={0,1}, lanes16-31 K={2,3}
      v2f a, b;
      a.x = As[wr + mi][kc + 2 * hi];
      a.y = As[wr + mi][kc + 2 * hi + 1];
      // B frag (4x16 f32): row K striped across lanes, same half-split
      b.x = Bs[wc + mi][kc + 2 * hi];
      b.y = Bs[wc + mi][kc + 2 * hi + 1];
      acc = __builtin_amdgcn_wmma_f32_16x16x4_f32(false, a, false, b,
                                                  (short)0, acc, false, false);
    }
    __syncthreads();
  }

  int col = colBase + wc + mi;
  if (col < n) {
    for (int v = 0; v < 8; ++v) {
      int row = rowBase + wr + v + 8 * hi;   // C/D layout: VGPR v -> M=v / v+8
      if (row < n) G[(size_t)row * ldG + col] = acc[v];
    }
  }
}

// ----------------------------- rsqrt of Gram diag --------------------------
__global__ __launch_bounds__(256)
void k_rnorm(const float* __restrict__ Gq, float* __restrict__ rn, int Q) {
  int q = blockIdx.x * 256 + threadIdx.x;
  if (q < Q) rn[q] = rsqrtf(fmaxf(Gq[(size_t)q * LDQ + q], 1e-30f));
}

// -------- per-class: members, exact median cos, pair count, kq -------------
__global__ __launch_bounds__(256)
void k_class_stats(const float* __restrict__ Gq, const float* __restrict__ rn,
                   const int* __restrict__ yboot, const float* __restrict__ distq,
                   const Scalars* __restrict__ sc,
                   int* __restrict__ clsM, float* __restrict__ clsTh,
                   int* __restrict__ clsPair, int* __restrict__ clsKq, int Q) {
  int c = blockIdx.x, tid = threadIdx.x;
  __shared__ int idxList[1024];
  __shared__ int mSh;
  __shared__ unsigned cntSh;
  __shared__ float loSh, hiSh;
  __shared__ float red[256];
  if (tid == 0) {
    int m = 0;
    if (c < sc->C) for (int q = 0; q < Q && m < 1024; ++q)
      if (yboot[q] == c) idxList[m++] = q;
    mSh = m;
  }
  __syncthreads();
  int m = mSh;
  if (c >= sc->C || m < 2) {
    if (tid == 0) { clsM[c] = m; clsPair[c] = 0; clsKq[c] = 0; clsTh[c] = 0.f; }
    return;
  }
  long total = (long)m * m;
  long r = (total - 1) / 2;                  // 0-based rank of the median
  if (tid == 0) { loSh = -4.f; hiSh = 4.f; }
  __syncthreads();
  for (int it = 0; it < 64; ++it) {
    float mid = 0.5f * (loSh + hiSh);
    if (tid == 0) cntSh = 0;
    __syncthreads();
    unsigned local = 0;
    for (long t = tid; t < total; t += 256) {
      int i = (int)(t / m), j = (int)(t % m);
      int qi = idxList[i], qj = idxList[j];
      float v = Gq[(size_t)qi * LDQ + qj] * rn[qi] * rn[qj];
      if (v <= mid) local++;
    }
    atomicAdd(&cntSh, local);
    __syncthreads();
    if (tid == 0) { if ((long)cntSh >= r + 1) hiSh = mid; else loSh = mid; }
    __syncthreads();
  }
  // exact median = smallest element strictly above lo
  float lo = loSh, lm = 1e30f;
  for (long t = tid; t < total; t += 256) {
    int i = (int)(t / m), j = (int)(t % m);
    int qi = idxList[i], qj = idxList[j];
    float v = Gq[(size_t)qi * LDQ + qj] * rn[qi] * rn[qj];
    if (v > lo && v < lm) lm = v;
  }
  red[tid] = lm; __syncthreads();
  for (int s = 128; s > 0; s >>= 1) {
    if (tid < s) red[tid] = fminf(red[tid], red[tid + s]);
    __syncthreads();
  }
  __shared__ float thSh;
  if (tid == 0) thSh = red[0];
  __syncthreads();
  float th = thSh;
  // count qualifying pairs i<j with cos >= thresh
  if (tid == 0) cntSh = 0;
  __syncthreads();
  unsigned local = 0;
  for (long t = tid; t < total; t += 256) {
    int i = (int)(t / m), j = (int)(t % m);
    if (i < j) {
      int qi = idxList[i], qj = idxList[j];
      float v = Gq[(size_t)qi * LDQ + qj] * rn[qi] * rn[qj];
      if (v >= th) local++;
    }
  }
  atomicAdd(&cntSh, local);
  __syncthreads();
  if (tid == 0) {
    clsM[c] = m; clsTh[c] = th; clsPair[c] = (int)cntSh;
    float best = 1e30f; int bi = idxList[0];
    for (int t = 0; t < m; ++t) {
      float v = distq[idxList[t] * CMAX + c];
      if (v < best) { best = v; bi = idxList[t]; }
    }
    clsKq[c] = bi;
  }
}

// ----------------- prefix offsets, nAll, k, flag resets --------------------
__global__ void k_scan(Scalars* sc, const int* __restrict__ clsM,
                       const int* __restrict__ clsPair, const int* __restrict__ clsKq,
                       int* __restrict__ clsBase, IPair* __restrict__ pairList, int Q) {
  int C = sc->C;
  int cursor = 0;
  for (int c = 0; c < C; ++c) {
    clsBase[c] = Q + cursor;
    if (clsM[c] >= 2) {
      cursor += clsPair[c];
      int slot = Q + cursor;                  // kq-virtual row after the pairs
      if (slot < MAX_N) { pairList[slot - Q].a = PROTO_FLAG | c; pairList[slot - Q].b = clsKq[c]; }
      cursor += 1;
    }
  }
  int n = Q + cursor; if (n > MAX_N) n = MAX_N;
  sc->nAll = n;
  int k = 31 - __clz(n);
  if (k < 1) k = 1; if (k > KMAX - 1) k = KMAX - 1;
  sc->kVal = k;
  sc->done = 0;
  sc->diffBits = 0u;
}

// --------------- deterministic emission of qualifying pairs ----------------
__global__ __launch_bounds__(256)
void k_emit_pairs(const float* __restrict__ Gq, const float* __restrict__ rn,
                  const int* __restrict__ yboot, const Scalars* __restrict__ sc,
                  const float* __restrict__ clsTh, const int* __restrict__ clsBase,
                  IPair* __restrict__ pairList, int Q) {
  int c = blockIdx.x, tid = threadIdx.x;
  __shared__ int idxList[1024];
  __shared__ int mSh, chunkTot, baseSh;
  __shared__ int flags[256], ranks[256];
  if (tid == 0) {
    int m = 0;
    if (c < sc->C) for (int q = 0; q < Q && m < 1024; ++q)
      if (yboot[q] == c) idxList[m++] = q;
    mSh = m;
    baseSh = clsBase[c];
  }
  __syncthreads();
  int m = mSh;
  if (c >= sc->C || m < 2) return;
  float th = clsTh[c];
  long P = (long)m * (m - 1) / 2;
  for (long s = 0; s < P; s += 256) {
    long p = s + tid;
    int qi = 0, qj = 0, flag = 0;
    if (p < P) {
      int i = 0; long rem = p;                 // canonical row-major triu order
      while (rem >= (long)(m - 1 - i)) { rem -= (m - 1 - i); i++; }
      int j = i + 1 + (int)rem;
      qi = idxList[i]; qj = idxList[j];
      float v = Gq[(size_t)qi * LDQ + qj] * rn[qi] * rn[qj];
      flag = (v >= th) ? 1 : 0;
    }
    flags[tid] = flag; __syncthreads();
    if (tid == 0) {
      int run = 0;
      for (int t = 0; t < 256; ++t) { ranks[t] = run; run += flags[t]; }
      chunkTot = run;
    }
    __syncthreads();
    if (flag) {
      int slot = baseSh + ranks[tid];
      if (slot < MAX_N) { pairList[slot - Q].a = qi; pairList[slot - Q].b = qj; }
    }
    __syncthreads();
    if (tid == 0) baseSh += chunkTot;
    __syncthreads();
  }
}

// ------------------------------ feat_all assembly --------------------------
__global__ __launch_bounds__(256)
void k_copyq(const float* __restrict__ fq, float* __restrict__ featAll) {
  int q = blockIdx.x;
  for (int d = threadIdx.x; d < D_FEAT; d += 256)
    featAll[(size_t)q * D_FEAT + d] = fq[(size_t)q * D_FEAT + d];
}

__global__ __launch_bounds__(256)
void k_fill_virt(const float* __restrict__ fq, const float* __restrict__ protos,
                 const Scalars* __restrict__ sc, const IPair* __restrict__ pairList,
                 float* __restrict__ featAll, int Q) {
  int s = blockIdx.x;
  int nv = sc->nAll - Q;
  if (s >= nv) return;
  IPair pr = pairList[s];
  const float* rowA = (pr.a & PROTO_FLAG) ? (protos + (size_t)(pr.a & 0xFFFF) * D_FEAT)
                                          : (fq + (size_t)pr.a * D_FEAT);
  const float* rowB = fq + (size_t)pr.b * D_FEAT;
  float* dst = featAll + (size_t)(Q + s) * D_FEAT;
  for (int d = threadIdx.x; d < D_FEAT; d += 256)
    dst[d] = 0.5f * (rowA[d] + rowB[d]);
}

// ------------------------- Gram -> pairwise distances ----------------------
__global__ __launch_bounds__(256)
void k_diag(const float* __restrict__ G, const Scalars* __restrict__ sc,
            float* __restrict__ diag) {
  int i = blockIdx.x * 256 + threadIdx.x;
  if (i < sc->nAll) diag[i] = G[(size_t)i * MAX_N + i];
}

__global__ __launch_bounds__(256)
void k_dist(float* __restrict__ G, const float* __restrict__ diag,
            const Scalars* __restrict__ sc) {
  int i = blockIdx.x;
  int j = blockIdx.y * 256 + threadIdx.x;
  int n = sc->nAll;
  if (i >= n || j >= n) return;
  size_t idx = (size_t)i * MAX_N + j;
  float d2 = diag[i] + diag[j] - 2.f * G[idx];
  d2 = fmaxf(d2, 0.f);
  G[idx] = sqrtf(fmaxf(d2, 1e-30f));
}

// --------------------------- per-row (k+1)-NN ------------------------------
__global__ __launch_bounds__(256)
void k_topk(const float* __restrict__ dist, const Scalars* __restrict__ sc,
            int* __restrict__ nbrs, float* __restrict__ sigma) {
  int i = blockIdx.x, tid = threadIdx.x;
  int n = sc->nAll;
  if (i >= n) return;
  int kk1 = sc->kVal + 1;
  __shared__ int chosen[KMAX];
  __shared__ float chosenD[KMAX];
  __shared__ float rv[256];
  __shared__ int ri[256];
  const float* row = dist + (size_t)i * MAX_N;
  for (int t = 0; t < kk1; ++t) {
    float bv = 1e30f; int bi = 0x7fffffff;
    for (int j = tid; j < n; j += 256) {
      bool ex = false;
      for (int u = 0; u < t; ++u) if (chosen[u] == j) { ex = true; break; }
      if (ex) continue;
      float v = row[j];
      if (v < bv || (v == bv && j < bi)) { bv = v; bi = j; }
    }
    rv[tid] = bv; ri[tid] = bi; __syncthreads();
    for (int s = 128; s > 0; s >>= 1) {
      if (tid < s) {
        float ov = rv[tid + s]; int oi = ri[tid + s];
        if (ov < rv[tid] || (ov == rv[tid] && oi < ri[tid])) { rv[tid] = ov; ri[tid] = oi; }
      }
      __syncthreads();
    }
    if (tid == 0) { chosen[t] = ri[0]; chosenD[t] = rv[0]; }
    __syncthreads();
  }
  if (tid == 0) {
    sigma[i] = chosenD[kk1 - 1] + 1e-8f;
    for (int t = 1; t < kk1; ++t) nbrs[(size_t)i * KMAX + (t - 1)] = chosen[t];
  }
}

// ------------------------------- W assembly --------------------------------
__global__ __launch_bounds__(256)
void k_zeroW(float* __restrict__ W) {
  size_t idx = ((size_t)blockIdx.x * 256 + threadIdx.x) * 4;
  float4 z = {0.f, 0.f, 0.f, 0.f};
  *(float4*)(W + idx) = z;
}

__global__ __launch_bounds__(256)
void k_scatterW(float* __restrict__ W, const float* __restrict__ dist,
                const int* __restrict__ nbrs, const float* __restrict__ sigma,
                const Scalars* __restrict__ sc) {
  int e = blockIdx.x * 256 + threadIdx.x;
  int i = e / KMAX, t = e % KMAX;
  if (i >= sc->nAll || t >= sc->kVal) return;
  int j = nbrs[(size_t)i * KMAX + t];
  float w = expf(-dist[(size_t)i * MAX_N + j] / (sigma[i] * sigma[j]));
  atomicAdd(&W[(size_t)i * MAX_N + j], 0.5f * w);  // each cell gets <=2 addends
  atomicAdd(&W[(size_t)j * MAX_N + i], 0.5f * w);  // -> order-invariant fp sum
}

__global__ __launch_bounds__(256)
void k_Dinv(const float* __restrict__ W, const Scalars* __restrict__ sc,
            float* __restrict__ Dinv) {
  __shared__ float red[256];
  int i = blockIdx.x, tid = threadIdx.x;
  int n = sc->nAll;
  if (i >= n) return;
  float s = 0.f;
  const float* Wr = W + (size_t)i * MAX_N;
  for (int j = tid; j < n; j += 256) s += Wr[j];
  red[tid] = s; __syncthreads();
  for (int t = 128; t > 0; t >>= 1) { if (tid < t) red[tid] += red[tid + t]; __syncthreads(); }
  if (tid == 0) Dinv[i] = 1.f / (red[0] + 1e-8f);
}

// --------------- a = cd2(feat_all, protos); d = cdist.min(1) ---------------
__global__ __launch_bounds__(256)
void k_protoDistAll(const float* __restrict__ featAll, const float* __restrict__ protos,
                    const Scalars* __restrict__ sc, float* __restrict__ aM,
                    float* __restrict__ dMin) {
  __shared__ float sred[256][CMAX];
  int i = blockIdx.x, tid = threadIdx.x;
  int n = sc->nAll;
  if (i >= n) return;
  int C = sc->C;
  float acc[CMAX];
  for (int c = 0; c < CMAX; ++c) acc[c] = 0.f;
  for (int d = tid; d < D_FEAT; d += 256) {
    float xv = featAll[(size_t)i * D_FEAT + d];
    for (int c = 0; c < C; ++c) {
      float df = xv - protos[c * D_FEAT + d];
      acc[c] += df * df;
    }
  }
  for (int c = 0; c < CMAX; ++c) sred[tid][c] = acc[c];
  __syncthreads();
  for (int s = 128; s > 0; s >>= 1) {
    if (tid < s) for (int c = 0; c < CMAX; ++c) sred[tid][c] += sred[tid + s][c];
    __syncthreads();
  }
  if (tid == 0) {
    float best = 1e30f;
    for (int c = 0; c < C; ++c) {
      float ss = fmaxf(sred[0][c], 0.f);
      aM[(size_t)i * CMAX + c] = ss;
      float dd = sqrtf(fmaxf(ss, 1e-30f));
      if (dd < best) best = dd;
    }
    dMin[i] = best;
  }
}

// --------------------------- exact median of dMin --------------------------
__global__ __launch_bounds__(256)
void k_medianD(const float* __restrict__ dMin, Scalars* sc) {
  __shared__ float loSh, hiSh;
  __shared__ unsigned cntSh;
  __shared__ float red[256];
  int tid = threadIdx.x;
  int n = sc->nAll;
  long r = (long)(n - 1) / 2;
  if (tid == 0) { loSh = -1.f; hiSh = 1e6f; }
  __syncthreads();
  for (int it = 0; it < 64; ++it) {
    float mid = 0.5f * (loSh + hiSh);
    if (tid == 0) cntSh = 0;
    __syncthreads();
    unsigned local = 0;
    for (int t = tid; t < n; t += 256) if (dMin[t] <= mid) local++;
    atomicAdd(&cntSh, local);
    __syncthreads();
    if (tid == 0) { if ((long)cntSh >= r + 1) hiSh = mid; else loSh = mid; }
    __syncthreads();
  }
  float lo = loSh, lm = 1e30f;
  for (int t = tid; t < n; t += 256) { float v = dMin[t]; if (v > lo && v < lm) lm = v; }
  red[tid] = lm; __syncthreads();
  for (int s = 128; s > 0; s >>= 1) { if (tid < s) red[tid] = fminf(red[tid], red[tid + s]); __syncthreads(); }
  if (tid == 0) { float med = red[0]; sc->lamDen = 2.f * med * med + 1e-8f; }
}

// ------------------------------- y0 and scale ------------------------------
__global__ __launch_bounds__(256)
void k_initY(const float* __restrict__ aM, const float* __restrict__ dMin,
             const float* __restrict__ Dinv, const Scalars* __restrict__ sc,
             float* __restrict__ y, float* __restrict__ scaleA) {
  int i = blockIdx.x * 256 + threadIdx.x;
  int n = sc->nAll;
  if (i >= n) return;
  int C = sc->C;
  float mx = -1e30f;
  for (int c = 0; c < C; ++c) mx = fmaxf(mx, -aM[(size_t)i * CMAX + c]);
  float sum = 0.f, e[CMAX];
  for (int c = 0; c < C; ++c) { e[c] = expf(-aM[(size_t)i * CMAX + c] - mx); sum += e[c]; }
  for (int c = 0; c < C; ++c) y[(size_t)i * CMAX + c] = e[c] / sum;
  float d = dMin[i];
  scaleA[i] = expf(-(d * d) / sc->lamDen) * Dinv[i];
}

// -------------------------- one propagation step ---------------------------
__global__ __launch_bounds__(256)
void k_step(const float* __restrict__ W, const float* __restrict__ y,
            float* __restrict__ yNew, const float* __restrict__ aM,
            const float* __restrict__ scaleA, Scalars* sc) {
  __shared__ float sred[256][CMAX];
  int i = blockIdx.x, tid = threadIdx.x;
  int n = sc->nAll;
  if (i >= n) return;
  if (sc->done) return;
  int C = sc->C;
  float acc[CMAX];
  for (int c = 0; c < CMAX; ++c) acc[c] = 0.f;
  const float* Wr = W + (size_t)i * MAX_N;
  for (int j = tid; j < n; j += 256) {
    float wv = Wr[j];
    if (wv != 0.f)
      for (int c = 0; c < C; ++c) acc[c] += wv * y[(size_t)j * CMAX + c];
  }
  for (int c = 0; c < CMAX; ++c) sred[tid][c] = acc[c];
  __syncthreads();
  for (int s = 128; s > 0; s >>= 1) {
    if (tid < s) for (int c = 0; c < CMAX; ++c) sred[tid][c] += sred[tid + s][c];
    __syncthreads();
  }
  if (tid == 0) {
    float si = scaleA[i];
    float logit[CMAX], mx = -1e30f;
    for (int c = 0; c < C; ++c) {
      logit[c] = -aM[(size_t)i * CMAX + c] + si * sred[0][c];
      mx = fmaxf(mx, logit[c]);
    }
    float sum = 0.f, e[CMAX];
    for (int c = 0; c < C; ++c) { e[c] = expf(logit[c] - mx); sum += e[c]; }
    float df = 0.f;
    for (int c = 0; c < C; ++c) {
      float yn = e[c] / sum;
      yNew[(size_t)i * CMAX + c] = yn;
      df = fmaxf(df, fabsf(yn - y[(size_t)i * CMAX + c]));
    }
    atomicMax(&sc->diffBits, __float_as_uint(df));  // df >= 0 -> bit-monotonic
  }
}

__global__ __launch_bounds__(256)
void k_commit(float* __restrict__ y, const float* __restrict__ yNew,
              const Scalars* __restrict__ sc) {
  int idx = blockIdx.x * 256 + threadIdx.x;
  if (idx >= sc->nAll * CMAX) return;
  if (!sc->done && __uint_as_float(sc->diffBits) >= 1e-4f) y[idx] = yNew[idx];
}

__global__ void k_flag(Scalars* sc) {
  float df = __uint_as_float(sc->diffBits);
  if (!sc->done && df < 1e-4f) sc->done = 1;
  sc->diffBits = 0u;
}

// ------------------------------- predictions -------------------------------
__global__ __launch_bounds__(256)
void k_preds(const float* __restrict__ y, const Scalars* __restrict__ sc,
             int* __restrict__ out, int Q) {
  int q = blockIdx.x * 256 + threadIdx.x;
  if (q >= Q) return;
  int C = sc->C;
  float best = -1e30f; int bi = 0;
  for (int c = 0; c < C; ++c) {
    float v = y[(size_t)q * CMAX + c];
    if (v > best) { best = v; bi = c; }      // strict > -> first max (argmax)
  }
  out[q] = bi;
}

// ---------------------------------------------------------------------------
extern "C" void kernel_launch(void* const* d_in, const int* in_sizes, int n_in,
                              void* d_out, int out_size, void* d_ws, size_t ws_size,
                              hipStream_t stream) {
  const float* feat_s = (const float*)d_in[0];
  const int*   y_s    = (const int*)d_in[1];
  const float* feat_q = (const float*)d_in[2];
  int S = in_sizes[0] / D_FEAT;
  int Q = in_sizes[2] / D_FEAT;
  (void)n_in; (void)out_size; (void)ws_size;   // needs ~560 MB workspace

  char* p = (char*)d_ws;
  auto alloc = [&](size_t bytes) -> void* {
    void* r = (void*)p;
    p += (bytes + 255) & ~(size_t)255;
    return r;
  };
  Scalars* sc     = (Scalars*)alloc(sizeof(Scalars));
  float*   protos = (float*)alloc(sizeof(float) * CMAX * D_FEAT);
  float*   distq  = (float*)alloc(sizeof(float) * (size_t)Q * CMAX);
  int*     yboot  = (int*)alloc(sizeof(int) * Q);
  float*   Gq     = (float*)alloc(sizeof(float) * (size_t)LDQ * LDQ);
  float*   rnormq = (float*)alloc(sizeof(float) * LDQ);
  int*     clsM   = (int*)alloc(sizeof(int) * CMAX);
  float*   clsTh  = (float*)alloc(sizeof(float) * CMAX);
  int*     clsPair= (int*)alloc(sizeof(int) * CMAX);
  int*     clsKq  = (int*)alloc(sizeof(int) * CMAX);
  int*     clsBase= (int*)alloc(sizeof(int) * CMAX);
  IPair*   pairL  = (IPair*)alloc(sizeof(IPair) * MAX_N);
  float*   featAll= (float*)alloc(sizeof(float) * (size_t)MAX_N * D_FEAT);
  float*   diag   = (float*)alloc(sizeof(float) * MAX_N);
  float*   sigma  = (float*)alloc(sizeof(float) * MAX_N);
  int*     nbrs   = (int*)alloc(sizeof(int) * (size_t)MAX_N * KMAX);
  float*   Dinv   = (float*)alloc(sizeof(float) * MAX_N);
  float*   aM     = (float*)alloc(sizeof(float) * (size_t)MAX_N * CMAX);
  float*   dMin   = (float*)alloc(sizeof(float) * MAX_N);
  float*   scaleA = (float*)alloc(sizeof(float) * MAX_N);
  float*   yCur   = (float*)alloc(sizeof(float) * (size_t)MAX_N * CMAX);
  float*   yNew   = (float*)alloc(sizeof(float) * (size_t)MAX_N * CMAX);
  float*   dist   = (float*)alloc(sizeof(float) * (size_t)MAX_N * MAX_N); // Gram then dist
  float*   W      = (float*)alloc(sizeof(float) * (size_t)MAX_N * MAX_N);

  k_protos<<<1, 256, 0, stream>>>(feat_s, y_s, protos, sc, S);
  k_distq<<<Q, 256, 0, stream>>>(feat_q, protos, sc, distq, yboot);

  dim3 gq((Q + 31) / 32, (Q + 31) / 32);
  k_gram_wmma<<<gq, 128, 0, stream>>>(feat_q, Gq, nullptr, Q, LDQ);
  k_rnorm<<<(Q + 255) / 256, 256, 0, stream>>>(Gq, rnormq, Q);

  k_class_stats<<<CMAX, 256, 0, stream>>>(Gq, rnormq, yboot, distq, sc,
                                          clsM, clsTh, clsPair, clsKq, Q);
  k_scan<<<1, 1, 0, stream>>>(sc, clsM, clsPair, clsKq, clsBase, pairL, Q);
  k_emit_pairs<<<CMAX, 256, 0, stream>>>(Gq, rnormq, yboot, sc, clsTh, clsBase, pairL, Q);
  k_copyq<<<Q, 256, 0, stream>>>(feat_q, featAll);
  k_fill_virt<<<MAX_N, 256, 0, stream>>>(feat_q, protos, sc, pairL, featAll, Q);

  dim3 ga(MAX_N / 32, MAX_N / 32);
  k_gram_wmma<<<ga, 128, 0, stream>>>(featAll, dist, sc, 0, MAX_N);
  k_diag<<<(MAX_N + 255) / 256, 256, 0, stream>>>(dist, sc, diag);
  dim3 gd(MAX_N, MAX_N / 256);
  k_dist<<<gd, 256, 0, stream>>>(dist, diag, sc);

  k_topk<<<MAX_N, 256, 0, stream>>>(dist, sc, nbrs, sigma);
  k_zeroW<<<(int)(((size_t)MAX_N * MAX_N) / 1024), 256, 0, stream>>>(W);
  k_scatterW<<<MAX_N * KMAX / 256, 256, 0, stream>>>(W, dist, nbrs, sigma, sc);
  k_Dinv<<<MAX_N, 256, 0, stream>>>(W, sc, Dinv);

  k_protoDistAll<<<MAX_N, 256, 0, stream>>>(featAll, protos, sc, aM, dMin);
  k_medianD<<<1, 256, 0, stream>>>(dMin, sc);
  k_initY<<<(MAX_N + 255) / 256, 256, 0, stream>>>(aM, dMin, Dinv, sc, yCur, scaleA);

  for (int it = 0; it < 50; ++it) {
    k_step<<<MAX_N, 256, 0, stream>>>(W, yCur, yNew, aM, scaleA, sc);
    k_commit<<<MAX_N * CMAX / 256, 256, 0, stream>>>(yCur, yNew, sc);
    k_flag<<<1, 1, 0, stream>>>(sc);
  }

  k_preds<<<(Q + 255) / 256, 256, 0, stream>>>(yCur, sc, (int*)d_out, Q);
}